// MeanAggregator_49280454754451
// MI455X (gfx1250) — compile-verified
//
#include <hip/hip_runtime.h>
#include <cstdint>

// MeanAggregator: out[b,d] = mean_k table[ids[b,k], d]
// B=65536, K=32, D=128. Memory-bound gather (~1.1 GB) -> target ~50us on MI455X.
//
// gfx1250 paths used:
//  - global_load_async_to_lds_b128 (32 outstanding async gathers per wave, no VGPR returns)
//  - s_wait_asynccnt
//  - v_wmma_f32_16x16x4_f32 for the K-reduction (A = uniform 1/32)

typedef __attribute__((ext_vector_type(2))) float v2f;
typedef __attribute__((ext_vector_type(8))) float v8f;

#define KSAMP 32
#define DIM 128
#define WAVES_PER_BLOCK 2

__global__ __launch_bounds__(WAVES_PER_BLOCK * 32)
void mean_agg_kernel(const float* __restrict__ table,
                     const int* __restrict__ ids,
                     float* __restrict__ out,
                     int nNodes)
{
    // Per-wave staging tile: 32 gathered rows of 128 f32 (16 KB) + 128 f32 result.
    __shared__ float stage[WAVES_PER_BLOCK][KSAMP][DIM];
    __shared__ float res[WAVES_PER_BLOCK][DIM];

    const int lane = threadIdx.x & 31;
    const int w    = threadIdx.x >> 5;
    const int node = blockIdx.x * WAVES_PER_BLOCK + w;
    if (node >= nNodes) return; // uniform per wave

    // Lane l owns gathered row ids[node*K + l] (coalesced 128B index load).
    const int idx = ids[node * KSAMP + lane];

    const uint64_t gaddr = (uint64_t)(uintptr_t)table + (uint64_t)(uint32_t)idx * (DIM * 4);
    const uint32_t laddr = (uint32_t)(uintptr_t)(&stage[w][lane][0]);

    // 32 async copies; instruction j moves bytes [16j, 16j+16) of every lane's
    // row. ISA: the immediate offset is added to BOTH the LDS and global
    // addresses, so one address VGPR (pair) per lane covers the whole row.
    #pragma unroll
    for (int j = 0; j < KSAMP; ++j) {
        asm volatile("global_load_async_to_lds_b128 %0, %1, off offset:%2"
                     :: "v"(laddr), "v"(gaddr), "i"(j * 16)
                     : "memory");
    }
    asm volatile("s_wait_asynccnt 0" ::: "memory");

    // K-reduction via WMMA f32 16x16x4: C[m,n] += sum_kk A[m,kk]*B[kk,n].
    // A is uniformly 1/32, so the (undocumented-order) mapping of the 4 K-rows
    // onto B's two VGPRs x two lane-halves is irrelevant: any assignment that
    // covers each row exactly once yields the same sum.
    const int n    = lane & 15;   // output column within the d-tile
    const int half = lane >> 4;
    const v2f a = {1.0f / KSAMP, 1.0f / KSAMP};

    #pragma unroll
    for (int t = 0; t < DIM / 16; ++t) {       // 8 d-tiles of 16
        v8f c = {};
        #pragma unroll
        for (int cc = 0; cc < KSAMP / 4; ++cc) { // 8 chunks of 4 k-samples
            const int k0 = cc * 4 + half * 2;
            v2f bv;
            bv.x = stage[w][k0 + 0][t * 16 + n];
            bv.y = stage[w][k0 + 1][t * 16 + n];
            // (neg_a, A, neg_b, B, c_mod, C, reuse_a, reuse_b)
            c = __builtin_amdgcn_wmma_f32_16x16x4_f32(
                    false, a, false, bv, (short)0, c, false, false);
        }
        // All 16 rows of C are identical; lanes 0..15, VGPR0 hold columns 0..15.
        if (lane < 16) res[w][t * 16 + n] = c[0];
    }

    // Same-wave LDS ordering (DScnt) makes res visible; emit one coalesced
    // 512B store per wave: lane l writes out[node, 4l..4l+4).
    float4 r = *(const float4*)&res[w][lane * 4];
    *(float4*)&out[(size_t)node * DIM + lane * 4] = r;
}

extern "C" void kernel_launch(void* const* d_in, const int* in_sizes, int n_in,
                              void* d_out, int out_size, void* d_ws, size_t ws_size,
                              hipStream_t stream)
{
    const float* table = (const float*)d_in[0];   // [E, 128] f32
    const int*   ids   = (const int*)d_in[1];     // [B, 32]  int
    float*       out   = (float*)d_out;           // [B, 128] f32

    const int nNodes = in_sizes[1] / KSAMP;
    const int blocks = (nNodes + WAVES_PER_BLOCK - 1) / WAVES_PER_BLOCK;

    hipLaunchKernelGGL(mean_agg_kernel,
                       dim3(blocks), dim3(WAVES_PER_BLOCK * 32), 0, stream,
                       table, ids, out, nNodes);
}